// gatNet_14224931685027
// MI455X (gfx1250) — compile-verified
//
#include <hip/hip_runtime.h>
#include <math.h>

typedef __attribute__((ext_vector_type(2))) float v2f;
typedef __attribute__((ext_vector_type(8))) float v8f;

#define NEG_ATTN 0.2f
#define NEG_ACT  0.01f

__device__ __forceinline__ float leaky(float x, float s) { return x > 0.f ? x : x * s; }

// Monotone float -> uint transform so unsigned atomicMax == float max.
__device__ __forceinline__ unsigned enc_f32(float f) {
    unsigned u = __float_as_uint(f);
    return (u & 0x80000000u) ? ~u : (u | 0x80000000u);
}
__device__ __forceinline__ float dec_f32(unsigned u) {
    return (u & 0x80000000u) ? __uint_as_float(u & 0x7FFFFFFFu) : __uint_as_float(~u);
}
#define ENC_NEG_INF 0x007FFFFFu  // enc(-inf)

// ---------------- Layer 1 GEMM: [N,3] x [3,64] -> [N,64], thread per (n,f)
__global__ void k_gemm_3x64(const float* __restrict__ x, const float* __restrict__ W,
                            float* __restrict__ h, int N) {
    int i = blockIdx.x * blockDim.x + threadIdx.x;
    if (i >= N * 64) return;
    int n = i >> 6, f = i & 63;
    const float* xr = x + n * 3;
    h[i] = xr[0] * W[f] + xr[1] * W[64 + f] + xr[2] * W[128 + f];
}

// ---------------- Layer 2 GEMM: [N,64] x [64,64] -> [N,64] via V_WMMA_F32_16X16X4_F32
__global__ void __launch_bounds__(128) k_gemm64_wmma(const float* __restrict__ x,
                                                     const float* __restrict__ W,
                                                     float* __restrict__ h, int N) {
    int lane = threadIdx.x & 31;
    int wave = threadIdx.x >> 5;
    int row0 = blockIdx.x * 16;
    if (row0 >= N) return;

    if (row0 + 16 > N) {  // scalar tail (not hit for N=100000)
        for (int idx = threadIdx.x; idx < (N - row0) * 64; idx += 128) {
            int r = row0 + (idx >> 6), c = idx & 63;
            float s = 0.f;
            for (int k = 0; k < 64; ++k) s += x[r * 64 + k] * W[k * 64 + c];
            h[r * 64 + c] = s;
        }
        return;
    }

    int col0 = wave * 16;
    int mi = lane & 15;   // M (for A) / N (for B,C,D)
    int kh = lane >> 4;   // which K-pair half
    const float* xr = x + (row0 + mi) * 64;

    v8f acc = {};
#pragma unroll
    for (int kt = 0; kt < 16; ++kt) {
        int k0 = kt * 4 + kh * 2;
        // A 16x4: VGPR0=K=k0, VGPR1=K=k0+1 for this lane's M
        v2f a = *(const v2f*)(xr + k0);
        // B 4x16: VGPR0=K=k0 row, VGPR1=K=k0+1 row, N=mi
        v2f b;
        b.x = W[k0 * 64 + col0 + mi];
        b.y = W[(k0 + 1) * 64 + col0 + mi];
        acc = __builtin_amdgcn_wmma_f32_16x16x4_f32(false, a, false, b,
                                                    (short)0, acc, false, false);
    }
#pragma unroll
    for (int v = 0; v < 8; ++v) {
        int row = row0 + v + 8 * kh;  // C/D layout: M = v + 8*(lane/16)
        h[row * 64 + col0 + mi] = acc[v];
    }
}

// ---------------- el/er for Fout=64: thread per node, float4 loads
__global__ void k_elr64(const float* __restrict__ h, const float* __restrict__ al,
                        const float* __restrict__ ar, float* __restrict__ el,
                        float* __restrict__ er, int N) {
    int n = blockIdx.x * blockDim.x + threadIdx.x;
    if (n >= N) return;
    const float4* hp = (const float4*)(h + n * 64);
    const float4* ap = (const float4*)al;
    const float4* rp = (const float4*)ar;
    float sl = 0.f, sr = 0.f;
#pragma unroll
    for (int c = 0; c < 16; ++c) {
        float4 hv = hp[c], a = ap[c], r = rp[c];
        sl += hv.x * a.x + hv.y * a.y + hv.z * a.z + hv.w * a.w;
        sr += hv.x * r.x + hv.y * r.y + hv.z * r.z + hv.w * r.w;
    }
    el[n] = sl;
    er[n] = sr;
}

// ---------------- Layer 3: fused [N,64]x[64,3] GEMM + el/er
__global__ void k_gemm64x3_elr(const float* __restrict__ x, const float* __restrict__ W,
                               const float* __restrict__ al, const float* __restrict__ ar,
                               float* __restrict__ h3, float* __restrict__ el,
                               float* __restrict__ er, int N) {
    int n = blockIdx.x * blockDim.x + threadIdx.x;
    if (n >= N) return;
    const float* xr = x + n * 64;
    float h0 = 0.f, h1 = 0.f, h2 = 0.f;
#pragma unroll 8
    for (int k = 0; k < 64; ++k) {
        float xv = xr[k];
        h0 += xv * W[k * 3 + 0];
        h1 += xv * W[k * 3 + 1];
        h2 += xv * W[k * 3 + 2];
    }
    h3[n * 3 + 0] = h0;
    h3[n * 3 + 1] = h1;
    h3[n * 3 + 2] = h2;
    el[n] = h0 * al[0] + h1 * al[1] + h2 * al[2];
    er[n] = h0 * ar[0] + h1 * ar[1] + h2 * ar[2];
}

// ---------------- init: out accumulator = 0, m = enc(-inf), denom = 0
__global__ void k_init(float* __restrict__ out, unsigned* __restrict__ m,
                       float* __restrict__ denom, int N, int Fout) {
    int i = blockIdx.x * blockDim.x + threadIdx.x;
    if (i < N * Fout) out[i] = 0.f;
    if (i < N) {
        m[i] = ENC_NEG_INF;
        denom[i] = 0.f;
    }
}

// ---------------- edge pass 1: logits + segment max (encoded uint atomicMax)
__global__ void k_edge_max(const float* __restrict__ el, const float* __restrict__ er,
                           const int* __restrict__ src, const int* __restrict__ dst,
                           float* __restrict__ ee, unsigned* __restrict__ m, int E) {
    int t = blockIdx.x * blockDim.x + threadIdx.x;
    if (t >= E) return;
    int s = src[t], d = dst[t];
    float v = leaky(el[s] + er[d], NEG_ATTN);
    ee[t] = v;
    atomicMax(&m[d], enc_f32(v));
}

// ---------------- edge pass 2: exp(e - m[dst]) + segment sum
__global__ void k_edge_expsum(float* __restrict__ ee, const unsigned* __restrict__ m,
                              float* __restrict__ denom, const int* __restrict__ dst, int E) {
    int t = blockIdx.x * blockDim.x + threadIdx.x;
    if (t >= E) return;
    int d = dst[t];
    float ex = __expf(ee[t] - dec_f32(m[d]));
    ee[t] = ex;
    atomicAdd(&denom[d], ex);
}

// ---------------- edge pass 3 (Fout=64): E*16 threads, float4 gather + 4 atomics
__global__ void k_scatter64(const float* __restrict__ h, const float* __restrict__ ee,
                            const float* __restrict__ denom, const int* __restrict__ src,
                            const int* __restrict__ dst, float* __restrict__ out, int E) {
    int t = blockIdx.x * blockDim.x + threadIdx.x;
    if (t >= E * 16) return;
    int e = t >> 4, c = t & 15;
    int s = src[e], d = dst[e];
    float a = ee[e] / denom[d];
    float4 hv = *(const float4*)(h + s * 64 + c * 4);
    float* o = out + d * 64 + c * 4;
    atomicAdd(o + 0, hv.x * a);
    atomicAdd(o + 1, hv.y * a);
    atomicAdd(o + 2, hv.z * a);
    atomicAdd(o + 3, hv.w * a);
}

// ---------------- edge pass 3 (Fout=3): thread per edge
__global__ void k_scatter3(const float* __restrict__ h, const float* __restrict__ ee,
                           const float* __restrict__ denom, const int* __restrict__ src,
                           const int* __restrict__ dst, float* __restrict__ out, int E) {
    int t = blockIdx.x * blockDim.x + threadIdx.x;
    if (t >= E) return;
    int s = src[t], d = dst[t];
    float a = ee[t] / denom[d];
    atomicAdd(out + d * 3 + 0, h[s * 3 + 0] * a);
    atomicAdd(out + d * 3 + 1, h[s * 3 + 1] * a);
    atomicAdd(out + d * 3 + 2, h[s * 3 + 2] * a);
}

// ---------------- bias + optional leaky_relu(0.01), in place
__global__ void k_bias_act(float* __restrict__ out, const float* __restrict__ b,
                           int N, int Fout, int act) {
    int i = blockIdx.x * blockDim.x + threadIdx.x;
    if (i >= N * Fout) return;
    float v = out[i] + b[i % Fout];
    out[i] = act ? leaky(v, NEG_ACT) : v;
}

static inline int gridFor(int n, int b) { return (n + b - 1) / b; }

extern "C" void kernel_launch(void* const* d_in, const int* in_sizes, int n_in,
                              void* d_out, int out_size, void* d_ws, size_t ws_size,
                              hipStream_t stream) {
    const float* feat = (const float*)d_in[0];
    const int* src = (const int*)d_in[1];
    const int* dst = (const int*)d_in[2];
    const float* W1 = (const float*)d_in[3];
    const float* al1 = (const float*)d_in[4];
    const float* ar1 = (const float*)d_in[5];
    const float* b1 = (const float*)d_in[6];
    const float* W2 = (const float*)d_in[7];
    const float* al2 = (const float*)d_in[8];
    const float* ar2 = (const float*)d_in[9];
    const float* b2 = (const float*)d_in[10];
    const float* W3 = (const float*)d_in[11];
    const float* al3 = (const float*)d_in[12];
    const float* ar3 = (const float*)d_in[13];
    const float* b3 = (const float*)d_in[14];

    const int N = in_sizes[0] / 3;
    const int E = in_sizes[1];

    // workspace carve-up (256B aligned)
    char* ws = (char*)d_ws;
    size_t off = 0;
    auto carve = [&](size_t bytes) {
        void* p = ws + off;
        off = (off + bytes + 255) & ~(size_t)255;
        return p;
    };
    float* bufA = (float*)carve((size_t)N * 64 * sizeof(float));  // h of current layer
    float* bufB = (float*)carve((size_t)N * 64 * sizeof(float));  // out accum / next x
    float* el = (float*)carve((size_t)N * sizeof(float));
    float* er = (float*)carve((size_t)N * sizeof(float));
    unsigned* m = (unsigned*)carve((size_t)N * sizeof(unsigned));
    float* denom = (float*)carve((size_t)N * sizeof(float));
    float* ee = (float*)carve((size_t)E * sizeof(float));

    const int B = 256;
    float* out3 = (float*)d_out;

    // ================= Layer 1: 3 -> 64 =================
    k_gemm_3x64<<<gridFor(N * 64, B), B, 0, stream>>>(feat, W1, bufA, N);
    k_elr64<<<gridFor(N, B), B, 0, stream>>>(bufA, al1, ar1, el, er, N);
    k_init<<<gridFor(N * 64, B), B, 0, stream>>>(bufB, m, denom, N, 64);
    k_edge_max<<<gridFor(E, B), B, 0, stream>>>(el, er, src, dst, ee, m, E);
    k_edge_expsum<<<gridFor(E, B), B, 0, stream>>>(ee, m, denom, dst, E);
    k_scatter64<<<gridFor(E * 16, B), B, 0, stream>>>(bufA, ee, denom, src, dst, bufB, E);
    k_bias_act<<<gridFor(N * 64, B), B, 0, stream>>>(bufB, b1, N, 64, 1);

    // ================= Layer 2: 64 -> 64 (WMMA GEMM) =================
    k_gemm64_wmma<<<gridFor(N, 16), 128, 0, stream>>>(bufB, W2, bufA, N);
    k_elr64<<<gridFor(N, B), B, 0, stream>>>(bufA, al2, ar2, el, er, N);
    k_init<<<gridFor(N * 64, B), B, 0, stream>>>(bufB, m, denom, N, 64);  // bufB reusable now
    k_edge_max<<<gridFor(E, B), B, 0, stream>>>(el, er, src, dst, ee, m, E);
    k_edge_expsum<<<gridFor(E, B), B, 0, stream>>>(ee, m, denom, dst, E);
    k_scatter64<<<gridFor(E * 16, B), B, 0, stream>>>(bufA, ee, denom, src, dst, bufB, E);
    k_bias_act<<<gridFor(N * 64, B), B, 0, stream>>>(bufB, b2, N, 64, 1);

    // ================= Layer 3: 64 -> 3 =================
    k_gemm64x3_elr<<<gridFor(N, B), B, 0, stream>>>(bufB, W3, al3, ar3, bufA, el, er, N);
    k_init<<<gridFor(N * 3, B), B, 0, stream>>>(out3, m, denom, N, 3);
    k_edge_max<<<gridFor(E, B), B, 0, stream>>>(el, er, src, dst, ee, m, E);
    k_edge_expsum<<<gridFor(E, B), B, 0, stream>>>(ee, m, denom, dst, E);
    k_scatter3<<<gridFor(E, B), B, 0, stream>>>(bufA, ee, denom, src, dst, out3, E);
    k_bias_act<<<gridFor(N * 3, B), B, 0, stream>>>(out3, b3, N, 3, 0);
}